// ClinicalEmbedding_68762426409609
// MI455X (gfx1250) — compile-verified
//
#include <hip/hip_runtime.h>
#include <stdint.h>

// ClinicalEmbedding: B=256, V=50, C=32, VOCAB=100000, E=128
// out[b,r,:] = 0                                        for r <  V-nv
//            = scale(idx)*W[idx], idx=flat[b, r-(V-nv)] for V-nv <= r < V-1
//            = sum_{j=nv-1}^{nv*C-1} scale*W[flat[b,j]] for r == V-1
// scale(v) = min(1, 1/max(||W[v]||, 1e-12))

typedef __attribute__((ext_vector_type(2))) float v2f;
typedef __attribute__((ext_vector_type(8))) float v8f;
typedef __attribute__((address_space(3))) int lds_int_t;

#define NB     256
#define NV     50
#define NC     32
#define NVOCAB 100000
#define NE     128
#define NL     (NV * NC)   // 1600 flattened codes per patient

// ---------------------------------------------------------------------------
// Kernel A: per-vocab-row renorm scale (one wave32 per row).
// ---------------------------------------------------------------------------
__global__ __launch_bounds__(256) void ce_scales(const float* __restrict__ W,
                                                 float* __restrict__ scales) {
  int row  = blockIdx.x * 8 + (threadIdx.x >> 5);
  int lane = threadIdx.x & 31;
  if (row >= NVOCAB) return;
  const float4* wr = (const float4*)(W + (size_t)row * NE);
  float4 t = wr[lane];                       // 32 lanes x 16B = 512B row
  float ss = t.x * t.x + t.y * t.y + t.z * t.z + t.w * t.w;
#pragma unroll
  for (int m = 16; m >= 1; m >>= 1) ss += __shfl_xor(ss, m, 32);
  float sc = fminf(1.0f, 1.0f / fmaxf(sqrtf(ss), 1e-12f));
  if (lane == 0) scales[row] = sc;
}

// ---------------------------------------------------------------------------
// Kernel B: pad-zero rows and single-lookup rows (r in [0, V-2]).
// One wave32 per output row, float4 coalesced.
// ---------------------------------------------------------------------------
__global__ __launch_bounds__(256) void ce_rows(const float* __restrict__ W,
                                               const int* __restrict__ codes,
                                               const int* __restrict__ nvis,
                                               const float* __restrict__ scales,
                                               float* __restrict__ out) {
  const int RPP = NV - 1;
  int gid  = blockIdx.x * 8 + (threadIdx.x >> 5);
  int lane = threadIdx.x & 31;
  if (gid >= NB * RPP) return;
  int b = gid / RPP;
  int r = gid - b * RPP;
  int nv = nvis[b];
  int i = r - (NV - nv);                     // bag index; <0 => pad. i<=nv-2 here.
  float4 val = make_float4(0.f, 0.f, 0.f, 0.f);
  if (i >= 0) {
    int idx   = codes[b * NL + i];
    float sc  = scales[idx];
    float4 t  = ((const float4*)(W + (size_t)idx * NE))[lane];
    val.x = t.x * sc; val.y = t.y * sc; val.z = t.z * sc; val.w = t.w * sc;
  }
  ((float4*)(out + ((size_t)b * NV + r) * NE))[lane] = val;
}

// ---------------------------------------------------------------------------
// Kernel C: last-bag sum via V_WMMA_F32_16X16X4_F32 with A = ones.
// D[m,n] = sum_k B[k,n] + C[m,n]  -> column sums, slot-permutation invariant.
// One workgroup (8 waves) per patient; wave w owns embedding columns
// [16w, 16w+16). Codes staged to LDS with async-to-LDS copies; main loop is
// branchless (clamp + select) with 16 rows / 4 WMMAs in flight per iteration.
// ---------------------------------------------------------------------------
__global__ __launch_bounds__(256) void ce_bag(const float* __restrict__ W,
                                              const int* __restrict__ codes,
                                              const int* __restrict__ nvis,
                                              const float* __restrict__ scales,
                                              float* __restrict__ out) {
  __shared__ int lcodes[NL];
  const int b    = blockIdx.x;
  const int tid  = threadIdx.x;
  const int wave = tid >> 5;                 // tile id 0..7
  const int lane = tid & 31;
  const int n    = lane & 15;                // column within tile
  const int h    = lane >> 4;                // lane half selects row pair
  const int* cb  = codes + b * NL;

  // Stage this patient's 1600 codes to LDS via CDNA5 async-to-LDS copies.
  for (int t = tid; t < NL; t += 256) {
    unsigned ldsaddr = (unsigned)(uintptr_t)(lds_int_t*)(&lcodes[t]);
    uint64_t gaddr   = (uint64_t)(uintptr_t)(cb + t);
    asm volatile("global_load_async_to_lds_b32 %0, %1, off"
                 :: "v"(ldsaddr), "v"(gaddr) : "memory");
  }
  asm volatile("s_wait_asynccnt 0" ::: "memory");
  __syncthreads();

  const int nv    = nvis[b];
  const int start = nv - 1;
  const int end   = nv * NC;                 // count = nv*31+1 >= 33
  const int ngroups = ((end - start) + 3) >> 2;
  const int col = wave * 16 + n;

  const v2f ones = {1.0f, 1.0f};
  v8f acc0 = {}, acc1 = {}, acc2 = {}, acc3 = {};

  for (int g = 0; g < ngroups; g += 4) {
    const int jbase = start + g * 4 + 2 * h;
    float vals[8];
#pragma unroll
    for (int u = 0; u < 4; ++u) {
      int ja = jbase + 4 * u;
      int jb = ja + 1;
      int ia = lcodes[min(ja, end - 1)];     // ds_load, clamped (branchless)
      int ib = lcodes[min(jb, end - 1)];
      int offa = ia * NE + col;              // 32-bit offsets -> saddr form
      int offb = ib * NE + col;
      float wa = W[offa] * scales[ia];
      float wb = W[offb] * scales[ib];
      vals[2 * u]     = (ja < end) ? wa : 0.f;
      vals[2 * u + 1] = (jb < end) ? wb : 0.f;
    }
    v2f b0 = {vals[0], vals[1]};
    v2f b1 = {vals[2], vals[3]};
    v2f b2 = {vals[4], vals[5]};
    v2f b3 = {vals[6], vals[7]};
    // EXEC is all-1s at the WMMAs (no divergent branches in this loop).
    acc0 = __builtin_amdgcn_wmma_f32_16x16x4_f32(false, ones, false, b0,
                                                 (short)0, acc0, false, false);
    acc1 = __builtin_amdgcn_wmma_f32_16x16x4_f32(false, ones, false, b1,
                                                 (short)0, acc1, false, false);
    acc2 = __builtin_amdgcn_wmma_f32_16x16x4_f32(false, ones, false, b2,
                                                 (short)0, acc2, false, false);
    acc3 = __builtin_amdgcn_wmma_f32_16x16x4_f32(false, ones, false, b3,
                                                 (short)0, acc3, false, false);
  }

  // Every D element in column n holds the column sum; take VGPR0.
  float s = (acc0[0] + acc1[0]) + (acc2[0] + acc3[0]);
  if (h == 0) out[((size_t)b * NV + (NV - 1)) * NE + col] = s;
}

// ---------------------------------------------------------------------------
extern "C" void kernel_launch(void* const* d_in, const int* in_sizes, int n_in,
                              void* d_out, int out_size, void* d_ws, size_t ws_size,
                              hipStream_t stream) {
  (void)in_sizes; (void)n_in; (void)out_size; (void)ws_size;
  const float* W      = (const float*)d_in[0];
  const int*   codes  = (const int*)d_in[1];
  const int*   nvis   = (const int*)d_in[2];
  float*       out    = (float*)d_out;
  float*       scales = (float*)d_ws;        // NVOCAB floats = 400 KB scratch

  ce_scales<<<(NVOCAB + 7) / 8, 256, 0, stream>>>(W, scales);
  ce_rows  <<<(NB * (NV - 1) + 7) / 8, 256, 0, stream>>>(W, codes, nvis, scales, out);
  ce_bag   <<<NB, 256, 0, stream>>>(W, codes, nvis, scales, out);
}